// TopKGating_3478923510213
// MI455X (gfx1250) — compile-verified
//
#include <hip/hip_runtime.h>
#include <math.h>

typedef float v2f __attribute__((ext_vector_type(2)));
typedef float v8f __attribute__((ext_vector_type(8)));

#define N_TOK 16384
#define DDIM  2048
#define NEXP  64
#define WAVES 8
#define ROWS_PER_BLOCK (WAVES * 16)   // 128 tokens per block
#define LSTRIDE 68                    // padded LDS row stride (floats)
#define NEG_INF (-3.402823466e38f)

// ---------------------------------------------------------------------------
// Zero the 129-float reduction scratch (64 p_sum, 64 freq, 1 lsesq).
// ---------------------------------------------------------------------------
__global__ void init_ws_kernel(float* __restrict__ ws) {
    int t = threadIdx.x;
    if (t < 129) ws[t] = 0.0f;
}

// ---------------------------------------------------------------------------
// Fused router: fp32 WMMA GEMM (x @ W^T) -> LDS -> top-2 / gates / lse /
// per-expert partial sums.
// ---------------------------------------------------------------------------
__global__ __launch_bounds__(256) void moe_router_kernel(
    const float* __restrict__ x,      // [N_TOK, DDIM]
    const float* __restrict__ W,      // [NEXP, DDIM]
    float* __restrict__ out,          // [4*N_TOK + 1]
    float* __restrict__ g_psum,       // [64]
    float* __restrict__ g_freq,       // [64]
    float* __restrict__ g_lsesq)      // [1]
{
    __shared__ float    s_logits[ROWS_PER_BLOCK * LSTRIDE];
    __shared__ float    s_lse[ROWS_PER_BLOCK];
    __shared__ float    s_lse2[ROWS_PER_BLOCK];
    __shared__ float    s_psum[4][NEXP];
    __shared__ unsigned s_freq[NEXP];

    const int tid  = threadIdx.x;
    const int wave = tid >> 5;
    const int lane = tid & 31;
    const int m    = lane & 15;   // M (A rows) / N (B experts) index
    const int half = lane >> 4;   // K sub-slot selector

    if (tid < NEXP) s_freq[tid] = 0u;

    const int row_base = blockIdx.x * ROWS_PER_BLOCK + wave * 16;

    // A-fragment stream: lane holds x[row_base+m][k + 2*half .. +1]
    const float* xp  = x + (size_t)(row_base + m) * DDIM + 2 * half;
    // B-fragment stream: lane holds W[16*t+m][k + 2*half .. +1]
    const float* wp0 = W + (size_t)m * DDIM + 2 * half;

    v8f acc[4];
    acc[0] = v8f{}; acc[1] = v8f{}; acc[2] = v8f{}; acc[3] = v8f{};

    for (int k0 = 0; k0 < DDIM; k0 += 16) {
        // stream x ahead (global_prefetch_b8)
        __builtin_prefetch(xp + k0 + 128, 0, 1);

        v2f a[4];
#pragma unroll
        for (int f = 0; f < 4; ++f)
            a[f] = *(const v2f*)(xp + k0 + 4 * f);

#pragma unroll
        for (int t = 0; t < 4; ++t) {
            const float* wp = wp0 + (size_t)t * 16 * DDIM + k0;
#pragma unroll
            for (int f = 0; f < 4; ++f) {
                v2f b = *(const v2f*)(wp + 4 * f);
                acc[t] = __builtin_amdgcn_wmma_f32_16x16x4_f32(
                    /*neg_a=*/false, a[f], /*neg_b=*/false, b,
                    /*c_mod=*/(short)0, acc[t],
                    /*reuse_a=*/false, /*reuse_b=*/false);
            }
        }
    }

    // Spill C tiles to LDS. VGPR j of a tile: lanes 0-15 hold M=j, lanes
    // 16-31 hold M=j+8; N = lane&15 (ISA 7.12.2).
#pragma unroll
    for (int t = 0; t < 4; ++t) {
#pragma unroll
        for (int j = 0; j < 8; ++j) {
            int r = wave * 16 + j + half * 8;
            s_logits[r * LSTRIDE + t * 16 + m] = acc[t][j];
        }
    }
    __syncthreads();

    // -------- per-row: top-2, gates, logsumexp --------
    if (tid < ROWS_PER_BLOCK) {
        const int r = tid;
        const size_t n = (size_t)blockIdx.x * ROWS_PER_BLOCK + r;
        const float* lrow = &s_logits[r * LSTRIDE];

        float mx = NEG_INF, v0 = NEG_INF, v1 = NEG_INF;
        int   i0 = 0, i1 = 0;
        for (int e = 0; e < NEXP; ++e) {
            float l = lrow[e];
            mx = fmaxf(mx, l);
            if (l > v0)      { v1 = v0; i1 = i0; v0 = l; i0 = e; }
            else if (l > v1) { v1 = l; i1 = e; }
        }
        float se = 0.0f;
        for (int e = 0; e < NEXP; ++e) se += expf(lrow[e] - mx);
        float lse = mx + logf(se);
        s_lse[r]  = lse;
        s_lse2[r] = lse * lse;

        float g0 = 1.0f / (1.0f + expf(v1 - v0));
        float g1 = 1.0f - g0;

        out[n * 2 + 0] = (float)i0;                  // top_k_indices
        out[n * 2 + 1] = (float)i1;
        out[(size_t)2 * N_TOK + n * 2 + 0] = g0;     // top_k_gates
        out[(size_t)2 * N_TOK + n * 2 + 1] = g1;

        atomicAdd(&s_freq[i0], 1u);
        atomicAdd(&s_freq[i1], 1u);
    }
    __syncthreads();

    // -------- per-expert probs column sums (all 256 threads) --------
    {
        const int e = tid & 63;
        const int g = tid >> 6;       // 4 row-groups of 32
        float p = 0.0f;
        const int r0 = g * 32;
        for (int r = r0; r < r0 + 32; ++r)
            p += expf(s_logits[r * LSTRIDE + e] - s_lse[r]);
        s_psum[g][e] = p;
    }
    __syncthreads();

    if (tid < NEXP) {
        float p = s_psum[0][tid] + s_psum[1][tid] + s_psum[2][tid] + s_psum[3][tid];
        atomicAdd(&g_psum[tid], p);
        atomicAdd(&g_freq[tid], (float)s_freq[tid]);
    }
    if (tid == 0) {
        float s = 0.0f;
        for (int r = 0; r < ROWS_PER_BLOCK; ++r) s += s_lse2[r];
        atomicAdd(g_lsesq, s);
    }
}

// ---------------------------------------------------------------------------
// Finalize: loss = E * sum(p_norm * f_norm) + 0.1 * lsesq / N
// ---------------------------------------------------------------------------
__global__ void finalize_kernel(const float* __restrict__ ws,
                                float* __restrict__ out) {
    __shared__ float sp[NEXP], sf[NEXP], spf[NEXP];
    const int t = threadIdx.x;
    float p = ws[t];
    float f = ws[64 + t];
    sp[t]  = fabsf(p);
    sf[t]  = fabsf(f);
    spf[t] = p * f;
    __syncthreads();
    for (int s = 32; s > 0; s >>= 1) {
        if (t < s) {
            sp[t]  += sp[t + s];
            sf[t]  += sf[t + s];
            spf[t] += spf[t + s];
        }
        __syncthreads();
    }
    if (t == 0) {
        float Sp = fmaxf(sp[0], 1e-12f);
        float Sf = fmaxf(sf[0], 1e-12f);
        float switchloss = (float)NEXP * spf[0] / (Sp * Sf);
        float z_loss = ws[128] / (float)N_TOK;
        out[(size_t)4 * N_TOK] = switchloss + 0.1f * z_loss;
    }
}

// ---------------------------------------------------------------------------
extern "C" void kernel_launch(void* const* d_in, const int* in_sizes, int n_in,
                              void* d_out, int out_size, void* d_ws, size_t ws_size,
                              hipStream_t stream) {
    const float* x = (const float*)d_in[0];
    const float* W = (const float*)d_in[1];
    float* out = (float*)d_out;
    float* ws  = (float*)d_ws;   // [0,64) p_sum, [64,128) freq, [128] lsesq

    init_ws_kernel<<<1, 256, 0, stream>>>(ws);
    moe_router_kernel<<<N_TOK / ROWS_PER_BLOCK, 256, 0, stream>>>(
        x, W, out, ws, ws + 64, ws + 128);
    finalize_kernel<<<1, 64, 0, stream>>>(ws, out);
}